// SAGE_6356551598791
// MI455X (gfx1250) — compile-verified
//
#include <hip/hip_runtime.h>
#include <hip/hip_bf16.h>
#include <stdint.h>

typedef float v2f __attribute__((ext_vector_type(2)));
typedef float v8f __attribute__((ext_vector_type(8)));

#define N_NODES 40000
#define FDIM    128
#define NCLS    16
#define PAD     132   // row pitch in floats: 528 B = 16 B aligned, bank-spread 4*row

// ---------------------------------------------------------------- utilities

__global__ void zero_f32(float* __restrict__ p, long long n) {
  long long i = (long long)blockIdx.x * blockDim.x + threadIdx.x;
  if (i < n) p[i] = 0.0f;
}

// degree (count of incoming edges per dst node)
__global__ void degree_kernel(const int* __restrict__ ei, float* __restrict__ cnt, int E) {
  int e = blockIdx.x * blockDim.x + threadIdx.x;
  if (e < E) atomicAdd(&cnt[ei[E + e]], 1.0f);
}

// scatter-accumulate: agg[dst, f] += xin[src, f]  (one thread per edge-feature).
// agg is L2-resident (20.5 MB << 192 MB L2) so these are L2-atomic RMWs.
__global__ void scatter_accum(const float* __restrict__ xin, const int* __restrict__ ei,
                              float* __restrict__ agg, int E) {
  long long idx = (long long)blockIdx.x * blockDim.x + threadIdx.x;
  if (idx >= (long long)E * FDIM) return;
  int e = (int)(idx >> 7);
  int f = (int)(idx & (FDIM - 1));
  int src = ei[e];
  int dst = ei[E + e];
  atomicAdd(&agg[(long long)dst * FDIM + f], xin[(long long)src * FDIM + f]);
}

// BN stats: relu then per-feature sum / sumsq; block-local accumulate, one
// atomic per feature per block.
__global__ __launch_bounds__(128) void bn_stats(const float* __restrict__ h,
                                                float* __restrict__ stats, int n) {
  int f = threadIdx.x;               // 0..127
  float s = 0.0f, s2 = 0.0f;
  for (int r = blockIdx.x; r < n; r += gridDim.x) {
    float v = fmaxf(h[(long long)r * FDIM + f], 0.0f);
    s += v; s2 += v * v;
  }
  atomicAdd(&stats[f], s);
  atomicAdd(&stats[FDIM + f], s2);
}

// BN apply (training-forward, biased variance), ReLU fused, in place.
__global__ void bn_apply(float* __restrict__ h, const float* __restrict__ stats,
                         const float* __restrict__ gamma, const float* __restrict__ beta,
                         int n) {
  long long idx = (long long)blockIdx.x * blockDim.x + threadIdx.x;
  if (idx >= (long long)n * FDIM) return;
  int f = (int)(idx & (FDIM - 1));
  float rn = 1.0f / (float)n;
  float mu  = stats[f] * rn;
  float var = stats[FDIM + f] * rn - mu * mu;
  float v = fmaxf(h[idx], 0.0f);
  h[idx] = (v - mu) * rsqrtf(var + 1e-5f) * gamma[f] + beta[f];
}

// ---------------------------------------------------------------- SAGE layer
//
// One wave per 16-row strip. The 16x128 agg-strip and x-strip are async-copied
// into LDS once (GLOBAL_LOAD_ASYNC_TO_LDS_B128, ASYNCcnt) with rows padded to
// 132 floats, so A-fragment reads are conflict-free ds loads instead of 512 B
// strided VMEM gathers repeated 8x. Then for each of 8 column tiles:
//   acc(16x16) = bias + sum_K (agg*1/deg)[16x4]@Wl^T[4x16] + x[16x4]@Wr^T[4x16]
// via v_wmma_f32_16x16x4_f32 (32+32 K-steps). Tile -> LDS, row L2-normalize,
// then either write H or run the fused classifier (one more WMMA chain).
//
// Fragment layouts (ISA 7.12.2, 32-bit):
//   A 16x4 : lanes 0-15 -> M=lane, VGPR0=K0 VGPR1=K1 ; lanes 16-31 -> K2,K3
//   B 4x16 : lanes 0-15 -> N=lane, rows K0/K1        ; lanes 16-31 -> rows K2/K3
//   C 16x16: VGPR v -> row v (lanes 0-15) / row v+8 (lanes 16-31), N = lane%16
__global__ __launch_bounds__(64) void sage_wmma(
    const float* __restrict__ agg, const float* __restrict__ cnt,
    const float* __restrict__ xin,
    const float* __restrict__ Wl, const float* __restrict__ bl,
    const float* __restrict__ Wr,
    float* __restrict__ hout,                 // do_fc == 0
    const float* __restrict__ Wfc, const float* __restrict__ bfc,
    float* __restrict__ fcout,                // do_fc == 1
    int do_fc)
{
  __shared__ __align__(16) float stage[2][2][16][PAD];  // [wave][agg/x][row][col]
  __shared__ __align__(16) float tile[2][16][PAD];
  const int wave  = threadIdx.x >> 5;
  const int lane  = threadIdx.x & 31;
  const int lhalf = lane & 15;
  const int khalf = (lane >> 4) << 1;         // 0 (lanes 0-15) or 2 (lanes 16-31)
  const int strip = blockIdx.x * 2 + wave;
  const int row0  = strip * 16;

  // warm the 128 KB weight working set into WGP$/L2
  for (int i = lane; i < FDIM; i += 32) {
    __builtin_prefetch(&Wl[i * FDIM], 0, 3);
    __builtin_prefetch(&Wr[i * FDIM], 0, 3);
  }

  // async-copy agg-strip and x-strip (16 rows x 512 B) into padded LDS rows
  {
    const unsigned lbA = (unsigned)(uintptr_t)&stage[wave][0][0][0];
    const unsigned lbX = (unsigned)(uintptr_t)&stage[wave][1][0][0];
    const unsigned long long gA =
        (unsigned long long)(uintptr_t)(agg + (size_t)row0 * FDIM);
    const unsigned long long gX =
        (unsigned long long)(uintptr_t)(xin + (size_t)row0 * FDIM);
    #pragma unroll
    for (int r = 0; r < 16; ++r) {
      unsigned long long ga = gA + (unsigned)(r * FDIM * 4 + lane * 16);
      unsigned la = lbA + (unsigned)(r * PAD * 4 + lane * 16);
      asm volatile("global_load_async_to_lds_b128 %0, %1, off"
                   :: "v"(la), "v"(ga) : "memory");
    }
    #pragma unroll
    for (int r = 0; r < 16; ++r) {
      unsigned long long ga = gX + (unsigned)(r * FDIM * 4 + lane * 16);
      unsigned la = lbX + (unsigned)(r * PAD * 4 + lane * 16);
      asm volatile("global_load_async_to_lds_b128 %0, %1, off"
                   :: "v"(la), "v"(ga) : "memory");
    }
    asm volatile("s_wait_asynccnt 0x0" ::: "memory");
  }

  const float invc = 1.0f / fmaxf(cnt[row0 + lhalf], 1.0f);

  for (int jb = 0; jb < 8; ++jb) {
    const int ncol = jb * 16 + lhalf;         // output column this lane covers
    const float bias = bl[ncol];
    v8f acc = {bias, bias, bias, bias, bias, bias, bias, bias};
    #pragma unroll 8
    for (int kk = 0; kk < FDIM; kk += 4) {    // mean(neigh) @ Wl^T
      v2f a, b;
      a.x = stage[wave][0][lhalf][kk + khalf]     * invc;
      a.y = stage[wave][0][lhalf][kk + khalf + 1] * invc;
      b.x = Wl[ncol * FDIM + kk + khalf];
      b.y = Wl[ncol * FDIM + kk + khalf + 1];
      acc = __builtin_amdgcn_wmma_f32_16x16x4_f32(false, a, false, b,
                                                  (short)0, acc, false, false);
    }
    #pragma unroll 8
    for (int kk = 0; kk < FDIM; kk += 4) {    // + x @ Wr^T
      v2f a, b;
      a.x = stage[wave][1][lhalf][kk + khalf];
      a.y = stage[wave][1][lhalf][kk + khalf + 1];
      b.x = Wr[ncol * FDIM + kk + khalf];
      b.y = Wr[ncol * FDIM + kk + khalf + 1];
      acc = __builtin_amdgcn_wmma_f32_16x16x4_f32(false, a, false, b,
                                                  (short)0, acc, false, false);
    }
    const int rb = (lane < 16) ? 0 : 8;
    #pragma unroll
    for (int v = 0; v < 8; ++v)
      tile[wave][rb + v][ncol] = acc[v];
  }
  __syncthreads();

  // row-wise L2 normalize: 2 lanes per row, 64 columns each, combine via shfl
  {
    const int r  = lane >> 1;
    const int c0 = (lane & 1) * 64;
    float ss = 0.0f;
    #pragma unroll 8
    for (int c = 0; c < 64; ++c) {
      float t = tile[wave][r][c0 + c];
      ss += t * t;
    }
    ss += __shfl_xor(ss, 1, 32);
    const float inv = 1.0f / fmaxf(sqrtf(ss), 1e-12f);
    #pragma unroll 8
    for (int c = 0; c < 64; ++c) {
      float t = tile[wave][r][c0 + c] * inv;
      tile[wave][r][c0 + c] = t;
      if (!do_fc) hout[(long long)(row0 + r) * FDIM + c0 + c] = t;
    }
  }
  __syncthreads();

  if (do_fc) {  // fused classifier: [16x128] @ Wfc^T[128x16] + bfc
    v8f acc = {};
    #pragma unroll 8
    for (int kk = 0; kk < FDIM; kk += 4) {
      v2f a, b;
      a.x = tile[wave][lhalf][kk + khalf];
      a.y = tile[wave][lhalf][kk + khalf + 1];
      b.x = Wfc[lhalf * FDIM + kk + khalf];
      b.y = Wfc[lhalf * FDIM + kk + khalf + 1];
      acc = __builtin_amdgcn_wmma_f32_16x16x4_f32(false, a, false, b,
                                                  (short)0, acc, false, false);
    }
    const float bias = bfc[lhalf];
    const int rb = (lane < 16) ? 0 : 8;
    #pragma unroll
    for (int v = 0; v < 8; ++v)
      fcout[(long long)(row0 + rb + v) * NCLS + lhalf] = acc[v] + bias;
  }
}

// ---------------------------------------------------------------- launcher

extern "C" void kernel_launch(void* const* d_in, const int* in_sizes, int n_in,
                              void* d_out, int out_size, void* d_ws, size_t ws_size,
                              hipStream_t stream) {
  const float* x     = (const float*)d_in[0];
  const int*   ei    = (const int*)d_in[1];
  const float* W1l   = (const float*)d_in[2];
  const float* b1l   = (const float*)d_in[3];
  const float* W1r   = (const float*)d_in[4];
  const float* gamma = (const float*)d_in[5];
  const float* beta  = (const float*)d_in[6];
  const float* W2l   = (const float*)d_in[7];
  const float* b2l   = (const float*)d_in[8];
  const float* W2r   = (const float*)d_in[9];
  const float* Wfc   = (const float*)d_in[10];
  const float* bfc   = (const float*)d_in[11];
  float* out = (float*)d_out;

  const int n = N_NODES;
  const int E = in_sizes[1] / 2;
  const long long NF = (long long)n * FDIM;

  // workspace: A (N*F) | H (N*F) | cnt (N) | stats (256)   ~41 MB
  float* A     = (float*)d_ws;
  float* H     = A + NF;
  float* cnt   = H + NF;
  float* stats = cnt + n;

  const unsigned gNF = (unsigned)((NF + 255) / 256);
  const unsigned gEF = (unsigned)(((long long)E * FDIM + 255) / 256);

  // ---- layer 1 ----
  zero_f32<<<gNF, 256, 0, stream>>>(A, NF);
  zero_f32<<<(n + 255) / 256, 256, 0, stream>>>(cnt, n);
  zero_f32<<<1, 256, 0, stream>>>(stats, 256);
  degree_kernel<<<(E + 255) / 256, 256, 0, stream>>>(ei, cnt, E);
  scatter_accum<<<gEF, 256, 0, stream>>>(x, ei, A, E);
  sage_wmma<<<n / 32, 64, 0, stream>>>(A, cnt, x, W1l, b1l, W1r,
                                       H, nullptr, nullptr, nullptr, 0);
  // ---- ReLU + BatchNorm (training forward, batch stats) ----
  bn_stats<<<256, 128, 0, stream>>>(H, stats, n);
  bn_apply<<<gNF, 256, 0, stream>>>(H, stats, gamma, beta, n);

  // ---- layer 2 + fused classifier ----
  zero_f32<<<gNF, 256, 0, stream>>>(A, NF);
  scatter_accum<<<gEF, 256, 0, stream>>>(H, ei, A, E);
  sage_wmma<<<n / 32, 64, 0, stream>>>(A, cnt, H, W2l, b2l, W2r,
                                       nullptr, Wfc, bfc, out, 1);
}